// DistanceAggregation_43473658970407
// MI455X (gfx1250) — compile-verified
//
#include <hip/hip_runtime.h>
#include <cstdint>

typedef float v2f __attribute__((ext_vector_type(2)));
typedef float v8f __attribute__((ext_vector_type(8)));
typedef unsigned int v4u __attribute__((ext_vector_type(4)));
typedef int v4i __attribute__((ext_vector_type(4)));
typedef int v8i __attribute__((ext_vector_type(8)));

constexpr int kT    = 1000;
constexpr int kD    = 512;
constexpr int kTM1  = 999;
constexpr int kStep = 15;   // scan steps per Gram block (rows 1..15; row 0 = rep)
constexpr int kNBlk = 67;   // 66*15 = 990 full + 9 tail

// ---- TDM: one instruction moves a 15x512 f32 tile global -> LDS (TENSORcnt) ----
// 2D descriptor: rows_valid limits tensor_dim1 so OOB tail rows read as zeros.
static __device__ __forceinline__ void tdm_load_tile(uint32_t lds_addr,
                                                     const float* gptr,
                                                     uint32_t rows_valid) {
  const uint64_t ga = (uint64_t)(uintptr_t)gptr;
  v4u g0;
  g0.x = 1u;                                               // count=1 (valid), load
  g0.y = lds_addr;                                         // LDS byte address
  g0.z = (uint32_t)ga;                                     // global_addr[31:0]
  g0.w = (uint32_t)((ga >> 32) & 0x1FFFFFFu) | (2u << 30); // addr[56:32] | type=2
  v8i g1;
  g1[0] = (int)(2u << 16);            // workgroup_mask=0, data_size=2 (4 bytes)
  g1[1] = (int)(512u << 16);          // tensor_dim0[15:0]=512 (bits 63:48)
  g1[2] = (int)(rows_valid << 16);    // tensor_dim0 hi=0 | tensor_dim1[15:0]
  g1[3] = (int)(512u << 16);          // tensor_dim1 hi=0 | tile_dim0=512
  g1[4] = kStep;                      // tile_dim1=15, tile_dim2=0 (2D)
  g1[5] = 512;                        // tensor_dim0_stride[31:0]=512
  g1[6] = 0;                          // stride hi | tensor_dim1_stride lo (unused)
  g1[7] = 0;
  const v4i gz4 = {0, 0, 0, 0};       // groups 2/3 unused (tile_dim2=0)
  const v8i gz8 = {0, 0, 0, 0, 0, 0, 0, 0};
  __builtin_amdgcn_tensor_load_to_lds(g0, g1, gz4, gz4, gz8, 0);
}

__global__ __launch_bounds__(32)
void seg_mean_scan(const float* __restrict__ data, const float* __restrict__ thr_p,
                   float* __restrict__ out_means, float* __restrict__ out_flags) {
  __shared__ float Xbuf[3][kStep][kD];  // 92160 B: triple-buffered frame tiles
  __shared__ float Rep[kD];             //  2048 B: current representative frame
  __shared__ float Gs[16 * 16];         //  1024 B: Gram matrix of the block

  const int   b    = blockIdx.x;
  const int   lane = threadIdx.x;
  const float thr  = thr_p[0];
  const float* dbase  = data + (size_t)b * kT * kD;
  float*       omeans = out_means + (size_t)b * kTM1 * kD;
  float*       oflags = out_flags + (size_t)b * kTM1;

  const uint32_t xb[3] = {(uint32_t)(uintptr_t)&Xbuf[0][0][0],
                          (uint32_t)(uintptr_t)&Xbuf[1][0][0],
                          (uint32_t)(uintptr_t)&Xbuf[2][0][0]};

  // ---- prologue: prefetch tiles for blocks 0 and 1 (frames 1..15, 16..30) ----
  tdm_load_tile(xb[0], dbase + (size_t)1 * kD, (uint32_t)(kT - 1));
  tdm_load_tile(xb[1], dbase + (size_t)(1 + kStep) * kD, (uint32_t)(kT - 1 - kStep));

  // segment running sum starts as frame 0 (segment = data[0:1]);
  // the same registers seed Rep (row 0 of every Gram until first boundary).
  float seg[16];
  {
    const float* f0 = dbase + lane * 16;
#pragma unroll
    for (int t = 0; t < 16; ++t) seg[t] = f0[t];
    float* rp = &Rep[lane * 16];
#pragma unroll
    for (int q = 0; q < 4; ++q) {
      float4 o;
      o.x = seg[4 * q + 0]; o.y = seg[4 * q + 1];
      o.z = seg[4 * q + 2]; o.w = seg[4 * q + 3];
      *(float4*)(rp + 4 * q) = o;
    }
  }
  int lenI = 1, fcnt = 0, rep_idx = 0;

  // WMMA operand addressing (A 16x4 f32 layout; symmetric product -> B == A regs)
  const int arow = lane & 15;            // logical matrix row handled by this lane
  const int ksel = (lane < 16) ? 0 : 2;  // VGPR0 holds K=0/2, VGPR1 K=1/3
  const int ncol = lane & 15;
  const int hi8  = (lane >> 4) * 8;      // C/D: lanes 16-31 hold M=8..15

  for (int blk = 0; blk < kNBlk; ++blk) {
    const int cur    = blk % 3;
    const int base_i = 1 + blk * kStep;

    // prefetch two blocks ahead (one TDM instruction)
    if (blk + 2 < kNBlk) {
      const int bn2 = base_i + 2 * kStep;
      tdm_load_tile(xb[(blk + 2) % 3], dbase + (size_t)bn2 * kD,
                    (uint32_t)(kT - bn2));
    }
    // retire everything except the two newest tiles -> tile `blk` is in LDS
    if (blk < kNBlk - 2) {
      asm volatile("s_wait_tensorcnt 2" ::: "memory");
    } else {
      asm volatile("s_wait_tensorcnt 0" ::: "memory");
    }
    // Rep writes (prologue ds_store or end-of-scan LDS copy) must be visible
    asm volatile("s_wait_dscnt 0" ::: "memory");

    // ---- Gram of [rep; f_i..f_{i+14}] via V_WMMA_F32_16X16X4_F32, K = 512 ----
    const float* myrow = (arow == 0) ? (const float*)&Rep[0]
                                     : (const float*)&Xbuf[cur][arow - 1][0];
    const float* ap = myrow + ksel;
    v8f acc = {0.f, 0.f, 0.f, 0.f, 0.f, 0.f, 0.f, 0.f};
#pragma unroll 8
    for (int c = 0; c < 128; ++c) {
      v2f abv = *(const v2f*)(ap + 4 * c);   // ds_load_b64, shared A==B operand
      acc = __builtin_amdgcn_wmma_f32_16x16x4_f32(
          false, abv, false, abv, (short)0, acc, false, false);
    }
#pragma unroll
    for (int v = 0; v < 8; ++v) Gs[(v + hi8) * 16 + ncol] = acc[v];
    asm volatile("s_wait_dscnt 0" ::: "memory");

    // ---- sequential scan over the block (uniform control flow) ----
    int rep_row = 0;  // rep is row 0 of this block's Gram until an in-block boundary
    const int rem    = kTM1 - blk * kStep;
    const int nsteps = rem < kStep ? rem : kStep;
    for (int j = 0; j < nsteps; ++j) {
      const int   il   = j + 1;
      const int   i    = base_i + j;
      const float g_rr = Gs[rep_row * 16 + rep_row];
      const float g_ii = Gs[il * 16 + il];
      const float g_ri = Gs[rep_row * 16 + il];
      const float d2   = g_rr + g_ii - 2.0f * g_ri;       // ||rep - frame_i||^2
      const float aclp = fminf((float)fcnt - 1.0f, 80.0f);
      const float fac  = 1.9f / (1.0f + expf(aclp)) + 0.4f;
      const float tf   = thr * fac;
      const bool  boundary = d2 > tf * tf;                // thr*fac > 0 always
      const float scale    = boundary ? (1.0f / (float)lenI) : 0.0f;

      float* orow = omeans + (size_t)(i - 1) * kD + lane * 16;
#pragma unroll
      for (int q = 0; q < 4; ++q) {
        float4 o;
        o.x = seg[4 * q + 0] * scale;
        o.y = seg[4 * q + 1] * scale;
        o.z = seg[4 * q + 2] * scale;
        o.w = seg[4 * q + 3] * scale;
        *(float4*)(orow + 4 * q) = o;
      }
      if (lane == 0) oflags[i - 1] = boundary ? 1.0f : 0.0f;

      if (boundary) {
        fcnt = 0; lenI = 0; rep_row = il; rep_idx = i;
#pragma unroll
        for (int t = 0; t < 16; ++t) seg[t] = 0.0f;
      } else {
        fcnt += 1;
      }
      const float* xrow = &Xbuf[cur][j][0] + lane * 16;   // frame i
#pragma unroll
      for (int t = 0; t < 16; ++t) seg[t] += xrow[t];
      lenI += 1;
    }

    // If a boundary happened in this block, the new rep is a row of the current
    // tile: refresh Rep with an LDS->LDS lane copy (no global reload, no TDM).
    if (rep_idx >= base_i) {
      const float* src = &Xbuf[cur][rep_idx - base_i][0] + lane * 16;
      float*       dst = &Rep[lane * 16];
#pragma unroll
      for (int q = 0; q < 4; ++q)
        *(float4*)(dst + 4 * q) = *(const float4*)(src + 4 * q);
    }
    // visibility handled by s_wait_dscnt at top of next iteration
  }
}

extern "C" void kernel_launch(void* const* d_in, const int* in_sizes, int n_in,
                              void* d_out, int out_size, void* d_ws, size_t ws_size,
                              hipStream_t stream) {
  const float* data = (const float*)d_in[0];
  const float* thr  = (const float*)d_in[1];
  const int B = in_sizes[0] / (kT * kD);
  float* out_means = (float*)d_out;
  float* out_flags = (float*)d_out + (size_t)B * kTM1 * kD;
  seg_mean_scan<<<dim3(B), dim3(32), 0, stream>>>(data, thr, out_means, out_flags);
}